// C_BatchNorm_14843406975464
// MI455X (gfx1250) — compile-verified
//
#include <hip/hip_runtime.h>
#include <stdint.h>

#define BATCH 32
#define TILE  256   // positions per block == blockDim.x

// Compute the fused per-position affine  o_i = A*z + c  from raw moments.
__device__ __forceinline__ void cbn_coeffs(
    float sx, float sy, float sxx, float sxy, float syy,
    float g00, float g01, float g10, float g11, float be0, float be1,
    float& a00, float& a01, float& a10, float& a11, float& c0, float& c1)
{
    const float invB   = 1.0f / (float)BATCH;
    const float invBm1 = 1.0f / (float)(BATCH - 1);
    const float mux = sx * invB;
    const float muy = sy * invB;
    // Unbiased covariance from raw moments: (Σzz' - B μμ') / (B-1)
    const float cxx = (sxx - sx * mux) * invBm1;
    const float cxy = (sxy - sx * muy) * invBm1;
    const float cyy = (syy - sy * muy) * invBm1;
    const float det = cxx * cyy - cxy * cxy;
    const float s   = sqrtf(det);
    const float t   = sqrtf(cxx + cyy + 2.0f * s);
    const float rt  = 1.0f / t;
    // A = gamma @ ((sigma + s*I) / t)
    a00 = (g00 * (cxx + s) + g01 * cxy) * rt;
    a01 = (g00 * cxy + g01 * (cyy + s)) * rt;
    a10 = (g10 * (cxx + s) + g11 * cxy) * rt;
    a11 = (g10 * cxy + g11 * (cyy + s)) * rt;
    // fold the mean subtraction and beta into the constant term
    c0 = be0 - a00 * mux - a01 * muy;
    c1 = be1 - a10 * mux - a11 * muy;
}

__global__ __launch_bounds__(TILE) void cbn_kernel(
    const float* __restrict__ z, const float* __restrict__ gamma,
    const float* __restrict__ beta, float* __restrict__ out, int N)
{
    __shared__ float smem[BATCH * TILE * 2];   // 64 KB staging tile

    const int tid = threadIdx.x;
    const int n0  = blockIdx.x * TILE;

    const float g00 = gamma[0], g01 = gamma[1], g10 = gamma[2], g11 = gamma[3];
    const float be0 = beta[0],  be1 = beta[1];

    if (n0 + TILE <= N) {
        // ---- Fast path: CDNA5 async global->LDS staging of z[0:32, n0:n0+256, :] ----
        // SADDR form: uniform 64-bit base in SGPRs, per-lane 32-bit byte offset.
        const uint64_t zbase   = (uint64_t)(uintptr_t)z + (uint64_t)n0 * 8ull;
        const uint32_t bstride = (uint32_t)N * 8u;                // bytes per batch slice
        const uint32_t lds0    = (uint32_t)(uintptr_t)(&smem[0]); // flat low 32 bits == LDS offset

        // unit u = tid + it*256 (16B units); b = u>>7, w = u&127.
        // voff = b*bstride + w*16; each iteration advances b by 2 -> two adds/iter.
        uint32_t voff = (uint32_t)(tid >> 7) * bstride + (uint32_t)((tid & 127) * 16);
        uint32_t la   = lds0 + (uint32_t)(tid * 16);
        #pragma unroll
        for (int it = 0; it < 16; ++it) {        // 4096 x 16B units / 256 threads
            asm volatile("global_load_async_to_lds_b128 %0, %1, %2"
                         :: "v"(la), "v"(voff), "s"(zbase) : "memory");
            voff += 2u * bstride;                // next pair of batch slices
            la   += 4096u;                       // 256 threads * 16B
        }
        asm volatile("s_wait_asynccnt 0" ::: "memory");
        __syncthreads();

        // ---- Pass 1: raw moments over the batch (from LDS) ----
        float sx = 0.f, sy = 0.f, sxx = 0.f, sxy = 0.f, syy = 0.f;
        #pragma unroll
        for (int b = 0; b < BATCH; ++b) {
            const float2 v = *(const float2*)&smem[(b * TILE + tid) * 2];
            sx += v.x; sy += v.y;
            sxx = fmaf(v.x, v.x, sxx);
            sxy = fmaf(v.x, v.y, sxy);
            syy = fmaf(v.y, v.y, syy);
        }
        float a00, a01, a10, a11, c0, c1;
        cbn_coeffs(sx, sy, sxx, sxy, syy, g00, g01, g10, g11, be0, be1,
                   a00, a01, a10, a11, c0, c1);

        // ---- Pass 2: apply fused affine (from LDS), coalesced float2 stores ----
        float* ob = out + ((uint64_t)n0 + (uint64_t)tid) * 2ull;
        #pragma unroll
        for (int b = 0; b < BATCH; ++b) {
            const float2 v = *(const float2*)&smem[(b * TILE + tid) * 2];
            float2 o;
            o.x = fmaf(a00, v.x, fmaf(a01, v.y, c0));
            o.y = fmaf(a10, v.x, fmaf(a11, v.y, c1));
            *(float2*)(ob + (uint64_t)b * (uint64_t)N * 2ull) = o;
        }
    } else {
        // ---- Tail path (not taken for the reference shape): direct global loads ----
        const int n = n0 + tid;
        if (n >= N) return;
        const float2* zf = (const float2*)z;
        float sx = 0.f, sy = 0.f, sxx = 0.f, sxy = 0.f, syy = 0.f;
        for (int b = 0; b < BATCH; ++b) {
            const float2 v = zf[(uint64_t)b * (uint64_t)N + (uint64_t)n];
            sx += v.x; sy += v.y;
            sxx = fmaf(v.x, v.x, sxx);
            sxy = fmaf(v.x, v.y, sxy);
            syy = fmaf(v.y, v.y, syy);
        }
        float a00, a01, a10, a11, c0, c1;
        cbn_coeffs(sx, sy, sxx, sxy, syy, g00, g01, g10, g11, be0, be1,
                   a00, a01, a10, a11, c0, c1);
        float2* of = (float2*)out;
        for (int b = 0; b < BATCH; ++b) {
            const float2 v = zf[(uint64_t)b * (uint64_t)N + (uint64_t)n];
            float2 o;
            o.x = fmaf(a00, v.x, fmaf(a01, v.y, c0));
            o.y = fmaf(a10, v.x, fmaf(a11, v.y, c1));
            of[(uint64_t)b * (uint64_t)N + (uint64_t)n] = o;
        }
    }
}

extern "C" void kernel_launch(void* const* d_in, const int* in_sizes, int n_in,
                              void* d_out, int out_size, void* d_ws, size_t ws_size,
                              hipStream_t stream) {
    const float* z     = (const float*)d_in[0];   // [B, C, H, W, 2] fp32
    const float* gamma = (const float*)d_in[1];   // [2, 2] fp32
    const float* beta  = (const float*)d_in[2];   // [2] fp32
    float* out = (float*)d_out;

    const int N = in_sizes[0] / (BATCH * 2);      // C*H*W positions
    const int grid = (N + TILE - 1) / TILE;
    hipLaunchKernelGGL(cbn_kernel, dim3(grid), dim3(TILE), 0, stream,
                       z, gamma, beta, out, N);
}